// GNNModel_3582002725150
// MI455X (gfx1250) — compile-verified
//
#include <hip/hip_runtime.h>
#include <hip/hip_bf16.h>

// ---------------------------------------------------------------------------
// GCN forward on MI455X (gfx1250).
//   * dense GEMMs (x@W1, h@W2): fp32 WMMA (V_WMMA_F32_16X16X4_F32), one wave32
//     per 16x16 tile, exact fp32 accuracy
//   * edge gather/scatter (the memory-bound hot path): b128 coalesced gathers
//     + global_atomic_add_f32 scatter
//   * self-loop + bias folded into the aggregation init to save a pass
// ---------------------------------------------------------------------------

typedef float v2f __attribute__((ext_vector_type(2)));
typedef float v8f __attribute__((ext_vector_type(8)));

#define N_NODES    100000
#define N_EDGES    3200000
#define F_IN       128
#define HDIM       64
#define NUM_GRAPHS 256

// ---------------------------------------------------------------------------
// Generic fill
// ---------------------------------------------------------------------------
__global__ void fill_kernel(float* __restrict__ p, float v, int n) {
    int i = blockIdx.x * blockDim.x + threadIdx.x;
    if (i < n) p[i] = v;
}

// deg[dst] += 1 per edge (deg pre-initialized to 1.0 for the self loop)
__global__ void degree_kernel(const int* __restrict__ dst,
                              float* __restrict__ deg, int e) {
    int i = blockIdx.x * blockDim.x + threadIdx.x;
    if (i < e) atomicAdd(&deg[dst[i]], 1.0f);
}

// deg -> d^-1/2 in place
__global__ void rsqrt_kernel(float* __restrict__ deg, int n) {
    int i = blockIdx.x * blockDim.x + threadIdx.x;
    if (i < n) {
        float d = deg[i];
        deg[i] = (d > 0.0f) ? __frsqrt_rn(d) : 0.0f;
    }
}

// ---------------------------------------------------------------------------
// C[n_rows x 64] = A[n_rows x K] * W[K x 64], fp32, row-major.
// One wave32 per 16x16 output tile, V_WMMA_F32_16X16X4_F32 over K/4 steps.
// n_rows % 16 == 0 (100000 = 6250*16).  Bounds guard is wave-uniform so EXEC
// stays all-1s around WMMA as the ISA requires.
// ---------------------------------------------------------------------------
__global__ void gemm_wmma_f32(const float* __restrict__ A,
                              const float* __restrict__ W,
                              float* __restrict__ C,
                              int n_rows, int K) {
    const int lane    = threadIdx.x & 31;
    const int waveid  = threadIdx.x >> 5;                  // 8 waves / block
    const int gwave   = blockIdx.x * 8 + waveid;
    const int tiles_n = HDIM / 16;                         // 4
    const int tile_m  = gwave / tiles_n;
    const int tile_n  = gwave % tiles_n;
    const int row0    = tile_m * 16;
    const int col0    = tile_n * 16;
    if (row0 >= n_rows) return;                            // wave-uniform

    const int m     = lane & 15;     // row of A tile / col of B tile
    const int khalf = lane >> 4;     // lanes 16-31 hold K+2/K+3 per ISA layout

    v8f acc = {};
    const float* arow = A + (size_t)(row0 + m) * K + 2 * khalf;
    const float* bcol = W + (size_t)(2 * khalf) * HDIM + col0 + m;
    for (int kb = 0; kb < K; kb += 4) {
        // A 16x4: VGPR0 = K+2*khalf, VGPR1 = K+2*khalf+1 (contiguous float2)
        v2f a;
        a.x = arow[kb];
        a.y = arow[kb + 1];
        // B 4x16: same K split across lane halves, N striped across lanes
        v2f b;
        b.x = bcol[(size_t)kb * HDIM];
        b.y = bcol[(size_t)(kb + 1) * HDIM];
        acc = __builtin_amdgcn_wmma_f32_16x16x4_f32(
            /*neg_a=*/false, a, /*neg_b=*/false, b,
            /*c_mod=*/(short)0, acc, /*reuse_a=*/false, /*reuse_b=*/false);
    }

    // C/D layout: VGPR j -> M = j (lanes 0-15) / M = 8+j (lanes 16-31), N = lane%16
    const int n     = lane & 15;
    const int mbase = khalf * 8;
    float* crow = C + (size_t)(row0 + mbase) * HDIM + col0 + n;
#pragma unroll
    for (int j = 0; j < 8; ++j)
        crow[(size_t)j * HDIM] = acc[j];
}

// ---------------------------------------------------------------------------
// agg init: agg[node][f] = hw[node][f] * dinv[node]^2 + bias[f]
// (self-loop contribution + bias, replaces a zero fill; float4 / b128 path)
// ---------------------------------------------------------------------------
__global__ void init_self_bias(const float* __restrict__ hw,
                               const float* __restrict__ dinv,
                               const float* __restrict__ bias,
                               float* __restrict__ agg, int n_rows) {
    int i = blockIdx.x * blockDim.x + threadIdx.x;      // over N * 16 quads
    if (i >= n_rows * (HDIM / 4)) return;
    int node = i >> 4;                                  // / (64/4)
    int q    = (i & 15) * 4;                            // feature quad
    float di = dinv[node];
    float d2 = di * di;
    const float4 h = *reinterpret_cast<const float4*>(hw + (size_t)node * HDIM + q);
    const float4 b = *reinterpret_cast<const float4*>(bias + q);
    float4 r;
    r.x = h.x * d2 + b.x;
    r.y = h.y * d2 + b.y;
    r.z = h.z * d2 + b.z;
    r.w = h.w * d2 + b.w;
    *reinterpret_cast<float4*>(agg + (size_t)node * HDIM + q) = r;
}

// ---------------------------------------------------------------------------
// Edge scatter: agg[dst] += hw[src] * dinv[src]*dinv[dst]
// 16 threads per edge, one float4 (b128) gather each; 16 consecutive lanes
// cover one fully-coalesced 256B row of hw[src] / agg[dst].
// ---------------------------------------------------------------------------
__global__ void scatter_edges(const int* __restrict__ src,
                              const int* __restrict__ dst,
                              const float* __restrict__ dinv,
                              const float* __restrict__ hw,
                              float* __restrict__ agg, int e) {
    long long t = (long long)blockIdx.x * blockDim.x + threadIdx.x;
    long long total = (long long)e * 16;
    if (t >= total) return;
    int ei = (int)(t >> 4);
    int c  = ((int)t & 15) * 4;
    int s  = src[ei];
    int d  = dst[ei];
    float nrm = dinv[s] * dinv[d];
    const float4 h = *reinterpret_cast<const float4*>(hw + (size_t)s * HDIM + c);
    float* ap = agg + (size_t)d * HDIM + c;
    atomicAdd(ap + 0, h.x * nrm);
    atomicAdd(ap + 1, h.y * nrm);
    atomicAdd(ap + 2, h.z * nrm);
    atomicAdd(ap + 3, h.w * nrm);
}

// In-place ReLU over agg (float4 / b128 path)
__global__ void relu_kernel(float* __restrict__ p, int n_quads) {
    int i = blockIdx.x * blockDim.x + threadIdx.x;
    if (i >= n_quads) return;
    float4 v = *reinterpret_cast<float4*>(p + (size_t)i * 4);
    v.x = fmaxf(v.x, 0.0f);
    v.y = fmaxf(v.y, 0.0f);
    v.z = fmaxf(v.z, 0.0f);
    v.w = fmaxf(v.w, 0.0f);
    *reinterpret_cast<float4*>(p + (size_t)i * 4) = v;
}

// pooled[batch[n]] += h[n]; counts[batch[n]] += 1
__global__ void pool_scatter(const float* __restrict__ h,
                             const int* __restrict__ batch,
                             float* __restrict__ pooled,
                             float* __restrict__ counts, int n_rows) {
    int i = blockIdx.x * blockDim.x + threadIdx.x;
    if (i >= n_rows * HDIM) return;
    int node = i >> 6;
    int f    = i & (HDIM - 1);
    int g    = batch[node];
    atomicAdd(&pooled[(size_t)g * HDIM + f], h[i]);
    if (f == 0) atomicAdd(&counts[g], 1.0f);
}

// out[g] = dot(pooled[g]/max(counts[g],1), Wfc) + bfc
__global__ void fc_kernel(const float* __restrict__ pooled,
                          const float* __restrict__ counts,
                          const float* __restrict__ Wfc,
                          const float* __restrict__ bfc,
                          float* __restrict__ out) {
    int g = blockIdx.x * blockDim.x + threadIdx.x;
    if (g >= NUM_GRAPHS) return;
    float cinv = 1.0f / fmaxf(counts[g], 1.0f);
    float s = 0.0f;
#pragma unroll
    for (int f = 0; f < HDIM; ++f)
        s += pooled[(size_t)g * HDIM + f] * cinv * Wfc[f];
    out[g] = s + bfc[0];
}

// ---------------------------------------------------------------------------
// Host-side orchestration (graph-capture safe: stream-only, ws scratch)
// ---------------------------------------------------------------------------
extern "C" void kernel_launch(void* const* d_in, const int* in_sizes, int n_in,
                              void* d_out, int out_size, void* d_ws, size_t ws_size,
                              hipStream_t stream) {
    (void)in_sizes; (void)n_in; (void)out_size; (void)ws_size;

    const float* x     = (const float*)d_in[0];
    const int*   eidx  = (const int*)d_in[1];     // [2, E]: src then dst
    const int*   batch = (const int*)d_in[2];
    const float* W1    = (const float*)d_in[3];
    const float* b1    = (const float*)d_in[4];
    const float* W2    = (const float*)d_in[5];
    const float* b2    = (const float*)d_in[6];
    const float* Wfc   = (const float*)d_in[7];
    const float* bfc   = (const float*)d_in[8];
    float*       out   = (float*)d_out;

    const int* src = eidx;
    const int* dst = eidx + N_EDGES;

    // Workspace layout (floats), 256B-aligned chunks
    float* ws = (float*)d_ws;
    const size_t OFF_DINV = 0;                                // N
    const size_t OFF_A    = 100032;                           // N*64 (hw)
    const size_t OFF_B    = OFF_A + (size_t)N_NODES * HDIM;   // N*64 (agg/h)
    const size_t OFF_POOL = OFF_B + (size_t)N_NODES * HDIM;   // 256*64
    const size_t OFF_CNT  = OFF_POOL + NUM_GRAPHS * HDIM;     // 256
    float* dinv   = ws + OFF_DINV;
    float* bufA   = ws + OFF_A;
    float* bufB   = ws + OFF_B;
    float* pooled = ws + OFF_POOL;
    float* counts = ws + OFF_CNT;

    const int NT = 256;
    const int NH  = N_NODES * HDIM;                           // 6.4M
    const int NQ  = NH / 4;                                   // 1.6M quads
    const int gemm_blocks = (N_NODES / 16) * (HDIM / 16) / 8; // 3125
    const long long scat_items = (long long)N_EDGES * 16;     // 51.2M
    const int scat_blocks = (int)((scat_items + NT - 1) / NT);

    // --- degree / normalization ---
    fill_kernel<<<(N_NODES + NT - 1) / NT, NT, 0, stream>>>(dinv, 1.0f, N_NODES);
    degree_kernel<<<(N_EDGES + NT - 1) / NT, NT, 0, stream>>>(dst, dinv, N_EDGES);
    rsqrt_kernel<<<(N_NODES + NT - 1) / NT, NT, 0, stream>>>(dinv, N_NODES);

    // --- layer 1:  h1 = relu(scatter(x@W1) + self + b1) ---
    gemm_wmma_f32<<<gemm_blocks, NT, 0, stream>>>(x, W1, bufA, N_NODES, F_IN);
    init_self_bias<<<(NQ + NT - 1) / NT, NT, 0, stream>>>(bufA, dinv, b1, bufB, N_NODES);
    scatter_edges<<<scat_blocks, NT, 0, stream>>>(src, dst, dinv, bufA, bufB, N_EDGES);
    relu_kernel<<<(NQ + NT - 1) / NT, NT, 0, stream>>>(bufB, NQ);

    // --- layer 2:  h2 = relu(scatter(h1@W2) + self + b2) ---
    gemm_wmma_f32<<<gemm_blocks, NT, 0, stream>>>(bufB, W2, bufA, N_NODES, HDIM);
    init_self_bias<<<(NQ + NT - 1) / NT, NT, 0, stream>>>(bufA, dinv, b2, bufB, N_NODES);
    scatter_edges<<<scat_blocks, NT, 0, stream>>>(src, dst, dinv, bufA, bufB, N_EDGES);
    relu_kernel<<<(NQ + NT - 1) / NT, NT, 0, stream>>>(bufB, NQ);

    // --- global mean pool + FC ---
    fill_kernel<<<(NUM_GRAPHS * HDIM + NT - 1) / NT, NT, 0, stream>>>(pooled, 0.0f, NUM_GRAPHS * HDIM);
    fill_kernel<<<1, NUM_GRAPHS, 0, stream>>>(counts, 0.0f, NUM_GRAPHS);
    pool_scatter<<<(NH + NT - 1) / NT, NT, 0, stream>>>(bufB, batch, pooled, counts, N_NODES);
    fc_kernel<<<1, NUM_GRAPHS, 0, stream>>>(pooled, counts, Wfc, bfc, out);
}